// GNN_91190745629253
// MI455X (gfx1250) — compile-verified
//
#include <hip/hip_runtime.h>
#include <hip/hip_bf16.h>

typedef __attribute__((ext_vector_type(2))) float v2f;
typedef __attribute__((ext_vector_type(8))) float v8f;

#define HID 64

__device__ __forceinline__ void atomAddF(float* p, float v) {
    unsafeAtomicAdd(p, v);   // hardware global_atomic_add_f32
}

// ---- init: deg=1 (self loop), pool sums/cnts = 0 -------------------------
__global__ void gnn_init_kernel(float* deg, float* psum, float* pcnt,
                                int n, int gsum, int g) {
    int i = blockIdx.x * blockDim.x + threadIdx.x;
    if (i < n)    deg[i]  = 1.0f;
    if (i < gsum) psum[i] = 0.0f;
    if (i < g)    pcnt[i] = 0.0f;
}

// ---- degree accumulation over col --------------------------------------
__global__ void gnn_deg_kernel(const int* __restrict__ col, float* deg, int nE) {
    int e = blockIdx.x * blockDim.x + threadIdx.x;
    if (e < nE) atomAddF(&deg[col[e]], 1.0f);
}

// ---- dinv = rsqrt(deg) in place ----------------------------------------
__global__ void gnn_rsqrt_kernel(float* deg, int n) {
    int i = blockIdx.x * blockDim.x + threadIdx.x;
    if (i < n) deg[i] = rsqrtf(deg[i]);
}

// ---- layer1 GEMM: g1 = dinv * (x @ W1); also seeds s1 (self loop) ------
__global__ void gnn_layer1_kernel(const float* __restrict__ x,
                                  const float* __restrict__ W1,
                                  const float* __restrict__ dinv,
                                  float* __restrict__ gA, float* __restrict__ sB,
                                  int n) {
    __shared__ float lw[6 * HID];
    __shared__ float lx[4 * 6];
    int node4 = blockIdx.x * 4;
    int ln = threadIdx.x >> 6;          // local node 0..3
    int ch = threadIdx.x & 63;
    if (threadIdx.x < 6 * HID) lw[threadIdx.x] = W1[threadIdx.x];
    if (threadIdx.x < 24) {
        int nn = threadIdx.x / 6, k = threadIdx.x % 6;
        int node = node4 + nn;
        lx[threadIdx.x] = (node < n) ? x[(long)node * 6 + k] : 0.0f;
    }
    __syncthreads();
    int node = node4 + ln;
    if (node >= n) return;
    float acc = 0.0f;
#pragma unroll
    for (int k = 0; k < 6; ++k) acc += lx[ln * 6 + k] * lw[k * HID + ch];
    float v = acc * dinv[node];
    long o = (long)node * HID + ch;
    gA[o] = v;
    sB[o] = v;
}

// ---- edge scatter: s[col] += g[row], one wave32 per edge ---------------
__global__ void gnn_scatter_kernel(const float* __restrict__ g,
                                   float* __restrict__ s,
                                   const int* __restrict__ row,
                                   const int* __restrict__ col, int nE) {
    int wid  = blockIdx.x * (blockDim.x >> 5) + (threadIdx.x >> 5);
    int lane = threadIdx.x & 31;
    if (wid >= nE) return;
    int r = row[wid];
    int c = col[wid];
    const float2 gv = *(const float2*)(g + (long)r * HID + lane * 2);
    float* dst = s + (long)c * HID + lane * 2;
    atomAddF(dst,     gv.x);
    atomAddF(dst + 1, gv.y);
}

// ---- finalize layer1: h1 = relu(dinv*s1 + b1) --------------------------
__global__ void gnn_fin1_kernel(const float* __restrict__ s1,
                                const float* __restrict__ dinv,
                                const float* __restrict__ b1,
                                float* __restrict__ h1, long total) {
    long i = (long)blockIdx.x * blockDim.x + threadIdx.x;
    if (i >= total) return;
    long node = i >> 6; int ch = (int)(i & 63);
    float v = dinv[node] * s1[i] + b1[ch];
    h1[i] = fmaxf(v, 0.0f);
}

// ---- layer2 GEMM via f32 WMMA: g2 = dinv * (h1 @ W2); seeds s2 ---------
// grid = nFull/32 blocks, 256 threads (8 waves): wave = (rblk 0..1, ntile 0..3)
__global__ void gnn_gemm2_wmma_kernel(const float* __restrict__ h1,
                                      const float* __restrict__ W2,
                                      const float* __restrict__ dinv,
                                      float* __restrict__ g2,
                                      float* __restrict__ s2) {
    __shared__ float lw[HID * HID];     // 16 KB
    for (int i = threadIdx.x; i < HID * HID; i += 256) lw[i] = W2[i];
    __syncthreads();

    int wave  = threadIdx.x >> 5;       // 0..7
    int lane  = threadIdx.x & 31;
    int ntile = wave & 3;               // output col tile (16 cols)
    int rblk  = wave >> 2;              // row half of the 32-node block
    long row0 = (long)blockIdx.x * 32 + rblk * 16;
    int  mrow  = lane & 15;
    int  khalf = lane >> 4;             // 0: K=4k..4k+1, 1: K=4k+2..4k+3
    int  colN  = ntile * 16 + mrow;

    const float* arow = h1 + (row0 + mrow) * HID;
    v8f acc = {};
#pragma unroll
    for (int kk = 0; kk < 16; ++kk) {
        int kbase = kk * 4 + khalf * 2;
        v2f a; a.x = arow[kbase];                  a.y = arow[kbase + 1];
        v2f b; b.x = lw[kbase * HID + colN];       b.y = lw[(kbase + 1) * HID + colN];
        acc = __builtin_amdgcn_wmma_f32_16x16x4_f32(
            false, a, false, b, (short)0, acc, false, false);
    }
    // C layout: VGPR j -> M = j + 8*khalf, N = colN
#pragma unroll
    for (int j = 0; j < 8; ++j) {
        long node = row0 + j + khalf * 8;
        float v = acc[j] * dinv[node];
        long o = node * HID + colN;
        g2[o] = v;
        s2[o] = v;
    }
}

// scalar remainder GEMM (only launched when n % 32 != 0; here it never is)
__global__ void gnn_gemm2_rem_kernel(const float* __restrict__ h1,
                                     const float* __restrict__ W2,
                                     const float* __restrict__ dinv,
                                     float* __restrict__ g2,
                                     float* __restrict__ s2,
                                     int base, int n) {
    int idx = blockIdx.x * blockDim.x + threadIdx.x;
    int node = base + (idx >> 6);
    int ch = idx & 63;
    if (node >= n) return;
    float acc = 0.0f;
    const float* arow = h1 + (long)node * HID;
    for (int k = 0; k < HID; ++k) acc += arow[k] * W2[k * HID + ch];
    float v = acc * dinv[node];
    long o = (long)node * HID + ch;
    g2[o] = v;
    s2[o] = v;
}

// ---- finalize layer2 + mean-pool accumulation --------------------------
__global__ void gnn_fin2_pool_kernel(const float* __restrict__ s2,
                                     const float* __restrict__ dinv,
                                     const float* __restrict__ b2,
                                     const int* __restrict__ batch,
                                     float* __restrict__ psum,
                                     float* __restrict__ pcnt, long total) {
    long i = (long)blockIdx.x * blockDim.x + threadIdx.x;
    if (i >= total) return;
    long node = i >> 6; int ch = (int)(i & 63);
    float v = fmaxf(dinv[node] * s2[i] + b2[ch], 0.0f);
    int g = batch[node];
    atomAddF(&psum[(long)g * HID + ch], v);
    if (ch == 0) atomAddF(&pcnt[g], 1.0f);
}

// ---- head: out[g] = relu(mean @ fw1 + fb1) @ fw2 + fb2 ; one wave/graph -
__global__ void gnn_head_kernel(const float* __restrict__ psum,
                                const float* __restrict__ pcnt,
                                const float* __restrict__ fw1,
                                const float* __restrict__ fb1,
                                const float* __restrict__ fw2,
                                const float* __restrict__ fb2,
                                float* __restrict__ out, int G) {
    int g = blockIdx.x * (blockDim.x >> 5) + (threadIdx.x >> 5);
    int lane = threadIdx.x & 31;
    if (g >= G) return;
    float inv = 1.0f / fmaxf(pcnt[g], 1.0f);
    float acc = fb1[lane];
    const float* ps = psum + (long)g * HID;
#pragma unroll 8
    for (int k = 0; k < HID; ++k) acc += (ps[k] * inv) * fw1[k * 32 + lane];
    acc = fmaxf(acc, 0.0f);
    float p = acc * fw2[lane];
    for (int off = 16; off > 0; off >>= 1) p += __shfl_down(p, off, 32);
    if (lane == 0) out[g] = p + fb2[0];
}

extern "C" void kernel_launch(void* const* d_in, const int* in_sizes, int n_in,
                              void* d_out, int out_size, void* d_ws, size_t ws_size,
                              hipStream_t stream) {
    const float* x     = (const float*)d_in[0];
    const int*   eidx  = (const int*)  d_in[1];
    // d_in[2] = edge_attr: unused by reference
    const int*   batch = (const int*)  d_in[3];
    const float* W1    = (const float*)d_in[4];
    const float* b1    = (const float*)d_in[5];
    const float* W2    = (const float*)d_in[6];
    const float* b2    = (const float*)d_in[7];
    const float* fw1   = (const float*)d_in[8];
    const float* fb1   = (const float*)d_in[9];
    const float* fw2   = (const float*)d_in[10];
    const float* fb2   = (const float*)d_in[11];
    float* out = (float*)d_out;

    const int N = in_sizes[0] / 6;       // 100000
    const int E = in_sizes[1] / 2;       // 1600000
    const int G = out_size;              // 1000
    const int* row = eidx;
    const int* col = eidx + E;

    // workspace layout (floats), 256-float aligned
    const long Npad = ((long)N + 255) & ~255L;
    float* ws   = (float*)d_ws;
    float* dinv = ws;                    // N (deg -> dinv in place)
    float* bufA = dinv + Npad;           // N*64 : g1, then h1
    float* bufB = bufA + Npad * HID;     // N*64 : s1, then g2
    float* bufC = bufB + Npad * HID;     // N*64 : s2
    float* psum = bufC + Npad * HID;     // G*64
    float* pcnt = psum + (long)G * HID;  // G
    (void)ws_size; (void)n_in;

    const long total = (long)N * HID;
    const int blkN   = (N + 255) / 256;
    const int blkE   = (E + 255) / 256;
    const int blkEw  = (E + 7) / 8;              // 8 waves/block, 1 edge/wave
    const int blkT   = (int)((total + 255) / 256);

    // degrees -> dinv
    gnn_init_kernel<<<blkN, 256, 0, stream>>>(dinv, psum, pcnt, N, G * HID, G);
    gnn_deg_kernel<<<blkE, 256, 0, stream>>>(col, dinv, E);
    gnn_rsqrt_kernel<<<blkN, 256, 0, stream>>>(dinv, N);

    // layer 1
    gnn_layer1_kernel<<<(N + 3) / 4, 256, 0, stream>>>(x, W1, dinv, bufA, bufB, N);
    gnn_scatter_kernel<<<blkEw, 256, 0, stream>>>(bufA, bufB, row, col, E);
    gnn_fin1_kernel<<<blkT, 256, 0, stream>>>(bufB, dinv, b1, bufA, total);   // h1 -> bufA

    // layer 2: WMMA GEMM (full 32-node blocks) + scalar remainder
    const int nFull = N & ~31;
    if (nFull > 0)
        gnn_gemm2_wmma_kernel<<<nFull / 32, 256, 0, stream>>>(bufA, W2, dinv, bufB, bufC);
    const int rem = N - nFull;
    if (rem > 0)
        gnn_gemm2_rem_kernel<<<(rem * HID + 255) / 256, 256, 0, stream>>>(
            bufA, W2, dinv, bufB, bufC, nFull, N);
    gnn_scatter_kernel<<<blkEw, 256, 0, stream>>>(bufB, bufC, row, col, E);

    // finalize + pool + head
    gnn_fin2_pool_kernel<<<blkT, 256, 0, stream>>>(bufC, dinv, b2, batch, psum, pcnt, total);
    gnn_head_kernel<<<(G + 7) / 8, 256, 0, stream>>>(psum, pcnt, fw1, fb1, fw2, fb2, out, G);
}